// Head_90984587199191
// MI455X (gfx1250) — compile-verified
//
#include <hip/hip_runtime.h>
#include <hip/hip_bf16.h>
#include <math.h>

// ---------------------------------------------------------------------------
// Problem constants (from reference)
// ---------------------------------------------------------------------------
#define BATCH 16
#define HH 56
#define WW 56
#define CC 256
#define HWPIX (HH * WW)          // 3136
#define IMG 224
#define RAD 16                   // round(4*sigma), sigma=4
#define KTAPS (2 * RAD + 1)      // 33

typedef __attribute__((ext_vector_type(2))) float v2f;
typedef __attribute__((ext_vector_type(8))) float v8f;

// LDS row stride for D (16 x 256 diff matrix), padded to kill bank conflicts:
// stride 260 -> lane n hits bank (4n + k) mod 64, distinct across 16 lanes.
#define DSTRIDE 260

// ---------------------------------------------------------------------------
// Kernel 1: per-pixel Mahalanobis distance via fp32 WMMA.
// One workgroup (256 thr = 8 waves) per pixel p.
// Wave w owns TWO row tiles of E = M @ D^T: rows [32w,32w+16) and
// [32w+16,32w+32).  Per k-step the B fragment (from LDS) is shared by both
// tiles, the A fragments stream from global in a 16-deep load clause, and two
// independent v_wmma_f32_16x16x4_f32 accumulator chains interleave.
// ---------------------------------------------------------------------------
__global__ __launch_bounds__(256) void mahal_wmma_kernel(
    const float* __restrict__ inputs,   // [B, HW, C]
    const float* __restrict__ mean,     // [HW, C]
    const float* __restrict__ cvar,     // [HW, C, C]
    float* __restrict__ mask56)         // [B, HW]
{
    __shared__ float Dlds[BATCH * DSTRIDE];
    __shared__ float quadLds[BATCH];

    const int p   = blockIdx.x;
    const int tid = threadIdx.x;

    // diff = inputs[b, p, :] - mean[p, :]  -> LDS (coalesced over c)
    #pragma unroll
    for (int e = tid; e < BATCH * CC; e += 256) {
        const int b = e >> 8;
        const int c = e & (CC - 1);
        Dlds[b * DSTRIDE + c] =
            inputs[((size_t)b * HWPIX + p) * CC + c] - mean[(size_t)p * CC + c];
    }
    if (tid < BATCH) quadLds[tid] = 0.0f;
    __syncthreads();

    const int wave = tid >> 5;       // 0..7 : owns rows [32*wave, 32*wave+32)
    const int lane = tid & 31;
    const int hi   = lane >> 4;      // 0/1 : K split within fragment
    const int lo   = lane & 15;      // A: row-in-tile ; B/C/D: batch column

    // A fragment sources: rows (32w + lo) and (32w + 16 + lo) of M
    const float* __restrict__ Arow0 =
        cvar + (size_t)p * (CC * CC) + (size_t)(wave * 32 + lo) * CC;
    const float* __restrict__ Arow1 = Arow0 + 16 * CC;
    const float* __restrict__ Brow  = &Dlds[lo * DSTRIDE];

    v8f acc0 = {};
    v8f acc1 = {};
    #pragma unroll 8
    for (int k0 = 0; k0 < CC; k0 += 4) {
        const int kk = k0 + (hi << 1);
        v2f a0 = *(const v2f*)(Arow0 + kk);    // global_load_b64, streamed
        v2f a1 = *(const v2f*)(Arow1 + kk);    // global_load_b64, streamed
        v2f b  = *(const v2f*)(Brow + kk);     // ds_load_b64, conflict-free
        acc0 = __builtin_amdgcn_wmma_f32_16x16x4_f32(
            false, a0, false, b, (short)0, acc0, false, false);
        acc1 = __builtin_amdgcn_wmma_f32_16x16x4_f32(
            false, a1, false, b, (short)0, acc1, false, false);
    }

    // acc0[j] = E[row = wave*32      + j + 8*hi][batch = lo]
    // acc1[j] = E[row = wave*32 + 16 + j + 8*hi][batch = lo]
    // partial quad for batch lo: sum_rows D[lo][row] * E[row][lo]
    const float* __restrict__ Dme = &Dlds[lo * DSTRIDE + wave * 32 + (hi << 3)];
    float partial = 0.0f;
    #pragma unroll
    for (int j = 0; j < 8; ++j) partial = fmaf(Dme[j],      acc0[j], partial);
    #pragma unroll
    for (int j = 0; j < 8; ++j) partial = fmaf(Dme[16 + j], acc1[j], partial);
    atomicAdd(&quadLds[lo], partial);   // ds_add_f32
    __syncthreads();

    if (tid < BATCH) {
        const float q = quadLds[tid];
        mask56[(size_t)tid * HWPIX + p] = sqrtf(fmaxf(q, 0.0f));
    }
}

// ---------------------------------------------------------------------------
// Kernel 0: init gaussian weights (normalized) + zero the 16 score slots.
// ---------------------------------------------------------------------------
__global__ void init_kernel(float* __restrict__ g, float* __restrict__ score)
{
    const int t = threadIdx.x;
    if (t < BATCH) score[t] = 0.0f;
    if (t == 0) {
        float tmp[KTAPS];
        float sum = 0.0f;
        for (int i = 0; i < KTAPS; ++i) {
            const float d = (float)(i - RAD);
            tmp[i] = expf(-(d * d) / 32.0f);   // 2*sigma^2 = 32
            sum += tmp[i];
        }
        for (int i = 0; i < KTAPS; ++i) g[i] = tmp[i] / sum;
    }
}

// ---------------------------------------------------------------------------
// Kernel 2: bilinear resize 56x56 -> 224x224 (half-pixel centers, edge clamp)
// ---------------------------------------------------------------------------
__global__ __launch_bounds__(256) void resize_kernel(
    const float* __restrict__ m56, float* __restrict__ out)
{
    const int idx = blockIdx.x * 256 + threadIdx.x;
    if (idx >= BATCH * IMG * IMG) return;
    const int x = idx % IMG;
    const int y = (idx / IMG) % IMG;
    const int b = idx / (IMG * IMG);

    const float sy = (y + 0.5f) * 0.25f - 0.5f;
    const float sx = (x + 0.5f) * 0.25f - 0.5f;
    int y0 = (int)floorf(sy);
    int x0 = (int)floorf(sx);
    const float wy = sy - (float)y0;
    const float wx = sx - (float)x0;
    int y1 = min(max(y0 + 1, 0), HH - 1);
    int x1 = min(max(x0 + 1, 0), WW - 1);
    y0 = min(max(y0, 0), HH - 1);
    x0 = min(max(x0, 0), WW - 1);

    const float* __restrict__ src = m56 + (size_t)b * HWPIX;
    const float v00 = src[y0 * WW + x0];
    const float v01 = src[y0 * WW + x1];
    const float v10 = src[y1 * WW + x0];
    const float v11 = src[y1 * WW + x1];
    out[idx] = (1.0f - wy) * ((1.0f - wx) * v00 + wx * v01) +
               wy * ((1.0f - wx) * v10 + wx * v11);
}

__device__ __forceinline__ int reflect101(int i) {
    i = (i < 0) ? -i : i;
    return (i > IMG - 1) ? (2 * (IMG - 1) - i) : i;
}

// ---------------------------------------------------------------------------
// Kernel 3: vertical 33-tap blur, reflect-101
// ---------------------------------------------------------------------------
__global__ __launch_bounds__(256) void blur_v_kernel(
    const float* __restrict__ in, float* __restrict__ out,
    const float* __restrict__ g)
{
    const int idx = blockIdx.x * 256 + threadIdx.x;
    if (idx >= BATCH * IMG * IMG) return;
    const int x = idx % IMG;
    const int y = (idx / IMG) % IMG;
    const int b = idx / (IMG * IMG);
    const float* __restrict__ src = in + (size_t)b * IMG * IMG;
    float s = 0.0f;
    #pragma unroll
    for (int t = -RAD; t <= RAD; ++t) {
        const int yy = reflect101(y + t);
        s = fmaf(g[t + RAD], src[yy * IMG + x], s);
    }
    out[idx] = s;
}

// ---------------------------------------------------------------------------
// Kernel 4: horizontal 33-tap blur, reflect-101; write final mask and
// block-max -> atomic per-image score (int-punned max, values >= 0).
// ---------------------------------------------------------------------------
__global__ __launch_bounds__(256) void blur_h_kernel(
    const float* __restrict__ in, float* __restrict__ mask,
    float* __restrict__ score, const float* __restrict__ g)
{
    __shared__ float red[256];
    const int idx = blockIdx.x * 256 + threadIdx.x;
    const int x = idx % IMG;
    const int y = (idx / IMG) % IMG;
    const int b = idx / (IMG * IMG);
    const float* __restrict__ src = in + ((size_t)b * IMG + y) * IMG;
    float s = 0.0f;
    #pragma unroll
    for (int t = -RAD; t <= RAD; ++t) {
        const int xx = reflect101(x + t);
        s = fmaf(g[t + RAD], src[xx], s);
    }
    mask[idx] = s;

    red[threadIdx.x] = s;
    __syncthreads();
    #pragma unroll
    for (int o = 128; o > 0; o >>= 1) {
        if (threadIdx.x < o)
            red[threadIdx.x] = fmaxf(red[threadIdx.x], red[threadIdx.x + o]);
        __syncthreads();
    }
    if (threadIdx.x == 0)
        atomicMax((int*)&score[b], __float_as_int(red[0]));
}

// ---------------------------------------------------------------------------
// Launcher
// ---------------------------------------------------------------------------
extern "C" void kernel_launch(void* const* d_in, const int* in_sizes, int n_in,
                              void* d_out, int out_size, void* d_ws, size_t ws_size,
                              hipStream_t stream)
{
    const float* inputs = (const float*)d_in[0];   // [16, 3136, 256]
    const float* mean   = (const float*)d_in[1];   // [3136, 256]
    const float* cvar   = (const float*)d_in[2];   // [3136, 256, 256]

    float* out   = (float*)d_out;
    float* score = out;                 // [16]
    float* maskO = out + BATCH;         // [16, 224, 224]

    float* wsf  = (float*)d_ws;
    float* g    = wsf;                                    // 64 floats reserved
    float* m56  = wsf + 64;                               // 16*3136
    float* bufA = m56 + BATCH * HWPIX;                    // 16*224*224
    float* bufB = bufA + BATCH * IMG * IMG;               // 16*224*224

    const int npix = BATCH * IMG * IMG;                   // 802816
    const int nblk = npix / 256;                          // 3136

    init_kernel<<<1, 64, 0, stream>>>(g, score);
    mahal_wmma_kernel<<<HWPIX, 256, 0, stream>>>(inputs, mean, cvar, m56);
    resize_kernel<<<nblk, 256, 0, stream>>>(m56, bufA);
    blur_v_kernel<<<nblk, 256, 0, stream>>>(bufA, bufB, g);
    blur_h_kernel<<<nblk, 256, 0, stream>>>(bufB, maskO, score, g);
}